// CKGAT_59708635349022
// MI455X (gfx1250) — compile-verified
//
#include <hip/hip_runtime.h>
#include <hip/hip_bf16.h>
#include <math.h>

typedef __attribute__((ext_vector_type(2))) float v2f;
typedef __attribute__((ext_vector_type(8))) float v8f;

#define B_ 512
#define T_ 32
#define N_ 8
#define D_ 64
#define ROWS (B_ * T_)   // 16384
#define NG_ 0.2f
#define WLD 66           // padded LDS pitch for transposed W (keeps b64 align + no bank conflicts)

__device__ __forceinline__ float elu1(float x) {
    return x > 0.0f ? x : expm1f(x);
}

// ---------------------------------------------------------------------------
// Kernel 1: v_k = W_GAT @ a_k  (k = 0,1,2), each v_k is a D-vector.
// pi factorization: (nh@W)·a1 + (nr@W)·a2 + (nt@W)·a3 = nh·v1 + nr·v2 + nt·v3
// ---------------------------------------------------------------------------
__global__ void ckgat_vcompute(const float* __restrict__ W,
                               const float* __restrict__ a,
                               float* __restrict__ vbuf) {
    int j = threadIdx.x;
    if (j >= 3 * D_) return;
    int k = j >> 6, d = j & 63;
    float s = 0.0f;
    #pragma unroll 8
    for (int c = 0; c < D_; ++c) s += W[d * D_ + c] * a[k * D_ + c];
    vbuf[j] = s;
}

// ---------------------------------------------------------------------------
// Kernel 2: base terms (also initializes accumulators, overwriting poison).
// e_u = mean_t ent[user_h0];  e_v = ent[items] + mean_t ent[item_h0]
// ---------------------------------------------------------------------------
__global__ void ckgat_init_base(const int* __restrict__ items,
                                const int* __restrict__ uh0,
                                const int* __restrict__ ih0,
                                const float* __restrict__ ent,
                                float* __restrict__ eu,
                                float* __restrict__ ev) {
    int idx = blockIdx.x * blockDim.x + threadIdx.x;
    if (idx >= B_ * D_) return;
    int b = idx >> 6, d = idx & 63;
    float su = 0.0f, sv = 0.0f;
    #pragma unroll 4
    for (int t = 0; t < T_; ++t) {
        su += ent[(size_t)uh0[b * T_ + t] * D_ + d];
        sv += ent[(size_t)ih0[b * T_ + t] * D_ + d];
    }
    eu[idx] = su * (1.0f / T_);
    ev[idx] = ent[(size_t)items[b] * D_ + d] + sv * (1.0f / T_);
}

// ---------------------------------------------------------------------------
// Kernel 3: attention + neighbor aggregation for one (side, layer).
// One wave32 per (b,t) row; lane owns dims {2*lane, 2*lane+1}.
// Writes x[row,:] = nei_rep + t_emb to xbuf.
// ---------------------------------------------------------------------------
__global__ void ckgat_attn(const int* __restrict__ nh,
                           const int* __restrict__ nr,
                           const int* __restrict__ nt,
                           const int* __restrict__ tix,
                           const float* __restrict__ ent,
                           const float* __restrict__ rel,
                           const float* __restrict__ vbuf,
                           float* __restrict__ xbuf) {
    int wave = threadIdx.x >> 5;
    int lane = threadIdx.x & 31;
    int row  = blockIdx.x * 8 + wave;
    int d0   = lane * 2;

    const float2* vp = (const float2*)vbuf;
    float2 v1 = vp[lane];
    float2 v2 = vp[32 + lane];
    float2 v3 = vp[64 + lane];

    const int* nhp = nh + row * N_;
    const int* nrp = nr + row * N_;
    const int* ntp = nt + row * N_;

    float2 hv[N_];
    float  pi[N_];
    #pragma unroll
    for (int n = 0; n < N_; ++n) {
        float2 h = *(const float2*)(ent + (size_t)nhp[n] * D_ + d0);
        float2 r = *(const float2*)(rel + (size_t)nrp[n] * D_ + d0);
        float2 t = *(const float2*)(ent + (size_t)ntp[n] * D_ + d0);
        hv[n] = h;
        float p = h.x * v1.x + h.y * v1.y
                + r.x * v2.x + r.y * v2.y
                + t.x * v3.x + t.y * v3.y;
        #pragma unroll
        for (int off = 16; off > 0; off >>= 1) p += __shfl_xor(p, off, 32);
        pi[n] = (p >= 0.0f) ? p : NG_ * p;   // LeakyReLU
    }

    // softmax over N=8 (every lane holds all pi after the reduction)
    float mx = pi[0];
    #pragma unroll
    for (int n = 1; n < N_; ++n) mx = fmaxf(mx, pi[n]);
    float ex[N_], s = 0.0f;
    #pragma unroll
    for (int n = 0; n < N_; ++n) { ex[n] = expf(pi[n] - mx); s += ex[n]; }
    float inv = 1.0f / s;

    float2 nei; nei.x = 0.0f; nei.y = 0.0f;
    #pragma unroll
    for (int n = 0; n < N_; ++n) {
        float w = ex[n] * inv;
        nei.x += w * hv[n].x;
        nei.y += w * hv[n].y;
    }
    float2 tv = *(const float2*)(ent + (size_t)tix[row] * D_ + d0);
    float2 x; x.x = nei.x + tv.x; x.y = nei.y + tv.y;
    *(float2*)(xbuf + (size_t)row * D_ + d0) = x;
}

// ---------------------------------------------------------------------------
// Kernel 4: y = elu(x @ W), summed over T, accumulated into e_side[b,:].
// One wave32 per b (32 rows = 2 M-tiles of 16). Uses V_WMMA_F32_16X16X4_F32.
// W is transposed into LDS (Wt[n][k], pitch 66) so every B fragment
// {W[k0][n], W[k0+1][n]} is a contiguous 8B-aligned ds_load_b64 straight into
// the even-aligned operand pair — no repacking movs.
// A layout (f32 16x4): lanes 0-15 -> M=lane, K={0,1}; lanes 16-31 -> K={2,3}.
// B layout (f32 4x16): lanes 0-15 -> N=lane, K={0,1}; lanes 16-31 -> K={2,3}.
// D layout: VGPR i -> M = i (+8 for lanes 16-31), N = lane&15.
// ---------------------------------------------------------------------------
__global__ void ckgat_wmma_elu_acc(const float* __restrict__ xbuf,
                                   const float* __restrict__ W,
                                   float* __restrict__ eacc) {
    __shared__ float Wt[D_ * WLD];
    // transposed cooperative load: global read coalesced, LDS store bank step 2
    for (int i = threadIdx.x; i < D_ * D_; i += 256) {
        int k = i >> 6, n = i & 63;
        Wt[n * WLD + k] = W[i];
    }
    __syncthreads();

    int wave = threadIdx.x >> 5;
    int lane = threadIdx.x & 31;
    int b    = blockIdx.x * 8 + wave;
    int m    = lane & 15;
    int koff = (lane >> 4) << 1;   // 0 for lanes 0-15, 2 for lanes 16-31

    float outsum0 = 0.0f, outsum1 = 0.0f, outsum2 = 0.0f, outsum3 = 0.0f;

    #pragma unroll
    for (int mt = 0; mt < 2; ++mt) {
        int row0 = b * T_ + mt * 16;
        const float* xrow = xbuf + (size_t)(row0 + m) * D_;
        v8f c0 = {}, c1 = {}, c2 = {}, c3 = {};
        #pragma unroll
        for (int kk = 0; kk < 16; ++kk) {
            int k0 = kk * 4 + koff;                 // even -> 8B aligned
            v2f a  = *(const v2f*)(xrow + k0);
            v2f b0 = *(const v2f*)(&Wt[(m     ) * WLD + k0]);
            v2f b1 = *(const v2f*)(&Wt[(m + 16) * WLD + k0]);
            v2f b2 = *(const v2f*)(&Wt[(m + 32) * WLD + k0]);
            v2f b3 = *(const v2f*)(&Wt[(m + 48) * WLD + k0]);
            c0 = __builtin_amdgcn_wmma_f32_16x16x4_f32(false, a, false, b0, (short)0, c0, false, false);
            c1 = __builtin_amdgcn_wmma_f32_16x16x4_f32(false, a, false, b1, (short)0, c1, false, false);
            c2 = __builtin_amdgcn_wmma_f32_16x16x4_f32(false, a, false, b2, (short)0, c2, false, false);
            c3 = __builtin_amdgcn_wmma_f32_16x16x4_f32(false, a, false, b3, (short)0, c3, false, false);
        }
        // ELU elementwise, then sum over the 16 rows (M) of this tile per column
        float s0 = 0.0f, s1 = 0.0f, s2 = 0.0f, s3 = 0.0f;
        #pragma unroll
        for (int i = 0; i < 8; ++i) {
            s0 += elu1(c0[i]); s1 += elu1(c1[i]); s2 += elu1(c2[i]); s3 += elu1(c3[i]);
        }
        s0 += __shfl_xor(s0, 16, 32);
        s1 += __shfl_xor(s1, 16, 32);
        s2 += __shfl_xor(s2, 16, 32);
        s3 += __shfl_xor(s3, 16, 32);
        outsum0 += s0; outsum1 += s1; outsum2 += s2; outsum3 += s3;
    }

    if (lane < 16) {   // race-free: exactly one wave owns b
        float* e = eacc + (size_t)b * D_;
        e[ 0 + lane] += outsum0;
        e[16 + lane] += outsum1;
        e[32 + lane] += outsum2;
        e[48 + lane] += outsum3;
    }
}

// ---------------------------------------------------------------------------
// Kernel 5: out[b] = sigmoid(dot(e_u[b], e_v[b])). One wave per b.
// ---------------------------------------------------------------------------
__global__ void ckgat_final(const float* __restrict__ eu,
                            const float* __restrict__ ev,
                            float* __restrict__ out) {
    int wave = threadIdx.x >> 5;
    int lane = threadIdx.x & 31;
    int b    = blockIdx.x * 8 + wave;
    float2 u = ((const float2*)(eu + (size_t)b * D_))[lane];
    float2 v = ((const float2*)(ev + (size_t)b * D_))[lane];
    float p = u.x * v.x + u.y * v.y;
    #pragma unroll
    for (int off = 16; off > 0; off >>= 1) p += __shfl_xor(p, off, 32);
    if (lane == 0) out[b] = 1.0f / (1.0f + expf(-p));
}

extern "C" void kernel_launch(void* const* d_in, const int* in_sizes, int n_in,
                              void* d_out, int out_size, void* d_ws, size_t ws_size,
                              hipStream_t stream) {
    const int*   items   = (const int*)d_in[0];
    const int*   user_h  = (const int*)d_in[1];
    const int*   user_t  = (const int*)d_in[3];
    const int*   user_nh = (const int*)d_in[4];
    const int*   user_nr = (const int*)d_in[5];
    const int*   user_nt = (const int*)d_in[6];
    const int*   item_h  = (const int*)d_in[7];
    const int*   item_t  = (const int*)d_in[9];
    const int*   item_nh = (const int*)d_in[10];
    const int*   item_nr = (const int*)d_in[11];
    const int*   item_nt = (const int*)d_in[12];
    const float* ent     = (const float*)d_in[13];
    const float* rel     = (const float*)d_in[14];
    const float* W       = (const float*)d_in[15];
    const float* aG      = (const float*)d_in[16];
    float*       out     = (float*)d_out;

    char* ws = (char*)d_ws;
    float* vbuf = (float*)ws;                                   // 192 floats
    float* eu   = (float*)(ws + 1024);                          // B*D
    float* ev   = (float*)(ws + 1024 + B_ * D_ * sizeof(float));
    float* xbuf = (float*)(ws + 1024 + 2 * B_ * D_ * sizeof(float)); // ROWS*D

    ckgat_vcompute<<<1, 192, 0, stream>>>(W, aG, vbuf);
    ckgat_init_base<<<(B_ * D_ + 255) / 256, 256, 0, stream>>>(
        items, user_h /* layer 0 slice */, item_h /* layer 0 slice */, ent, eu, ev);

    for (int side = 0; side < 2; ++side) {
        const int* nh = side == 0 ? user_nh : item_nh;
        const int* nr = side == 0 ? user_nr : item_nr;
        const int* nt = side == 0 ? user_nt : item_nt;
        const int* tx = side == 0 ? user_t  : item_t;
        float*     e  = side == 0 ? eu : ev;
        for (int layer = 0; layer < 2; ++layer) {
            const int* nhL = nh + (size_t)layer * ROWS * N_;
            const int* nrL = nr + (size_t)layer * ROWS * N_;
            const int* ntL = nt + (size_t)layer * ROWS * N_;
            const int* txL = tx + (size_t)layer * ROWS;
            ckgat_attn<<<ROWS / 8, 256, 0, stream>>>(nhL, nrL, ntL, txL,
                                                     ent, rel, vbuf, xbuf);
            ckgat_wmma_elu_acc<<<B_ / 8, 256, 0, stream>>>(xbuf, W, e);
        }
    }

    ckgat_final<<<B_ / 8, 256, 0, stream>>>(eu, ev, out);
}